// Net_14173392077423
// MI455X (gfx1250) — compile-verified
//
#include <hip/hip_runtime.h>
#include <hip/hip_bf16.h>
#include <math.h>

// Problem constants (match reference)
static constexpr int NT  = 32768;   // total nodes
static constexpr int B_  = 16;      // graphs
static constexpr int N_  = 2048;    // nodes per graph
static constexpr int CIN = 64;
static constexpr int HID = 128;
static constexpr int K_  = 32;
static constexpr int E_  = 524288;  // edges
#define GN_EPS 1e-5f

typedef float v2f __attribute__((ext_vector_type(2)));
typedef float v8f __attribute__((ext_vector_type(8)));

__device__ __forceinline__ v8f wmma4(v2f a, v2f b, v8f c) {
  // V_WMMA_F32_16X16X4_F32 : D = A(16x4,f32) * B(4x16,f32) + C(16x16,f32)
  return __builtin_amdgcn_wmma_f32_16x16x4_f32(false, a, false, b, (short)0, c,
                                               false, false);
}

__device__ __forceinline__ float selu_f(float x) {
  const float alpha = 1.6732632423543772848f;
  const float scale = 1.0507009873554804934f;
  return x > 0.f ? scale * x : scale * alpha * (expf(x) - 1.f);
}

// ---------------- GraphNorm ----------------
__global__ void k_mean(const float* __restrict__ x, float* __restrict__ mean) {
  __shared__ float lds[256];
  int b = blockIdx.x;
  int c = threadIdx.x & 63;
  int g = threadIdx.x >> 6;                 // 4 node-groups
  const float* xb = x + (size_t)b * N_ * CIN;
  float s = 0.f;
  for (int n = g; n < N_; n += 4) s += xb[n * CIN + c];
  lds[threadIdx.x] = s;
  __syncthreads();
  if (g == 0) {
    float t = lds[c] + lds[c + 64] + lds[c + 128] + lds[c + 192];
    mean[b * CIN + c] = t / (float)N_;
  }
}

__global__ void k_var(const float* __restrict__ x, const float* __restrict__ mean,
                      const float* __restrict__ ms, float* __restrict__ var) {
  __shared__ float lds[256];
  int b = blockIdx.x;
  int c = threadIdx.x & 63;
  int g = threadIdx.x >> 6;
  const float* xb = x + (size_t)b * N_ * CIN;
  float mu = ms[c] * mean[b * CIN + c];
  float s = 0.f;
  for (int n = g; n < N_; n += 4) {
    float d = xb[n * CIN + c] - mu;
    s += d * d;
  }
  lds[threadIdx.x] = s;
  __syncthreads();
  if (g == 0) {
    float t = lds[c] + lds[c + 64] + lds[c + 128] + lds[c + 192];
    var[b * CIN + c] = t / (float)N_;
  }
}

__global__ void k_norm(const float* __restrict__ x, const float* __restrict__ mean,
                       const float* __restrict__ var, const float* __restrict__ ms,
                       const float* __restrict__ gw, const float* __restrict__ gb,
                       float* __restrict__ h) {
  int i = blockIdx.x * 256 + threadIdx.x;   // NT*CIN
  int c = i & 63;
  int n = i >> 6;
  int b = n >> 11;                          // n / 2048
  float xm = x[i] - ms[c] * mean[b * CIN + c];
  h[i] = gw[c] * xm * rsqrtf(var[b * CIN + c] + GN_EPS) + gb[c];
}

// ---------------- degree / init ----------------
__global__ void k_init(float* __restrict__ t, float* __restrict__ deg,
                       float* __restrict__ degcnt) {
  int i = blockIdx.x * 256 + threadIdx.x;   // covers NT*K
  t[i] = 0.f;
  if (i < NT) { deg[i] = 1.f; degcnt[i] = 0.f; }  // deg starts at 1 (self loop)
}

__global__ void k_degree(const int* __restrict__ ei, float* __restrict__ deg,
                         float* __restrict__ degcnt) {
  int e = blockIdx.x * 256 + threadIdx.x;
  if (e >= E_) return;
  atomicAdd(deg + ei[E_ + e], 1.f);    // in-degree on dst (for GCN norm)
  atomicAdd(degcnt + ei[e], 1.f);      // out-degree on src (adj row sums)
}

__global__ void k_dinv(float* __restrict__ deg) {
  int i = blockIdx.x * 256 + threadIdx.x;
  if (i < NT) deg[i] = rsqrtf(deg[i]);      // deg >= 1 always
}

// ---------------- GEMM1: hw = h(NT x 64) @ w1(64 x 128), WMMA f32 ----------------
__global__ void k_gemm1(const float* __restrict__ h, const float* __restrict__ w1,
                        float* __restrict__ hw) {
  int lane = threadIdx.x & 31;
  int wave = threadIdx.x >> 5;              // 8 waves -> 8 N-tiles (128 cols)
  int half = lane >> 4, idx = lane & 15;
  int m0 = blockIdx.x << 4;
  int n0 = wave << 4;
  const float* A = h + (size_t)(m0 + idx) * CIN;
  v8f acc = {};
  for (int k = 0; k < CIN; k += 4) {
    v2f a, b;
    a.x = A[k + 2 * half];
    a.y = A[k + 2 * half + 1];
    b.x = w1[(k + 2 * half) * HID + n0 + idx];
    b.y = w1[(k + 2 * half + 1) * HID + n0 + idx];
    acc = wmma4(a, b, acc);
  }
#pragma unroll
  for (int v = 0; v < 8; ++v)
    hw[(size_t)(m0 + v + 8 * half) * HID + n0 + idx] = acc[v];
}

// ---------------- GCN aggregation ----------------
__global__ void k_selfloop(const float* __restrict__ hw, const float* __restrict__ dinv,
                           float* __restrict__ agg) {
  int i = blockIdx.x * 256 + threadIdx.x;   // NT*HID
  int n = i >> 7;
  float dv = dinv[n];
  agg[i] = dv * dv * hw[i];
}

__global__ void k_edgeagg(const int* __restrict__ ei, const float* __restrict__ hw,
                          const float* __restrict__ dinv, float* __restrict__ agg) {
  int t = blockIdx.x * 256 + threadIdx.x;   // E*HID lanes
  int e = t >> 7;
  int c = t & 127;
  int s = ei[e], d = ei[E_ + e];
  float nrm = dinv[s] * dinv[d];
  atomicAdd(agg + (size_t)d * HID + c, nrm * hw[(size_t)s * HID + c]);
}

__global__ void k_selu1(float* __restrict__ agg, const float* __restrict__ b1) {
  int i = blockIdx.x * 256 + threadIdx.x;   // NT*HID, in-place -> h2
  int c = i & 127;
  agg[i] = selu_f(agg[i] + b1[c]);
}

// ---------------- GEMM2: logits = h2(NT x 128) @ w2(128 x 32) + b2 ----------------
__global__ void k_gemm2(const float* __restrict__ h2, const float* __restrict__ w2,
                        const float* __restrict__ b2, float* __restrict__ logits) {
  int lane = threadIdx.x & 31;
  int wave = threadIdx.x >> 5;
  int half = lane >> 4, idx = lane & 15;
  int m0 = (blockIdx.x * 4 + (wave >> 1)) << 4;
  int n0 = (wave & 1) << 4;
  const float* A = h2 + (size_t)(m0 + idx) * HID;
  v8f acc = {};
  for (int k = 0; k < HID; k += 4) {
    v2f a, b;
    a.x = A[k + 2 * half];
    a.y = A[k + 2 * half + 1];
    b.x = w2[(k + 2 * half) * K_ + n0 + idx];
    b.y = w2[(k + 2 * half + 1) * K_ + n0 + idx];
    acc = wmma4(a, b, acc);
  }
  float bias = b2[n0 + idx];
#pragma unroll
  for (int v = 0; v < 8; ++v)
    logits[(size_t)(m0 + v + 8 * half) * K_ + n0 + idx] = acc[v] + bias;
}

// ---------------- softmax over K=32 per node -> s ----------------
__global__ void k_softmax(const float* __restrict__ logits, float* __restrict__ s) {
  int n = blockIdx.x * 256 + threadIdx.x;
  if (n >= NT) return;
  const float* L = logits + (size_t)n * K_;
  float mx = L[0];
  for (int k = 1; k < K_; ++k) mx = fmaxf(mx, L[k]);
  float ev[K_];
  float sum = 0.f;
  for (int k = 0; k < K_; ++k) { ev[k] = expf(L[k] - mx); sum += ev[k]; }
  float inv = 1.f / sum;
  for (int k = 0; k < K_; ++k) s[(size_t)n * K_ + k] = ev[k] * inv;
}

// ---------------- GEMM3: out = selu(s^T @ xd) per graph, WMMA ----------------
// A[m=cluster, kk=node] = s[node, cluster] (transposed gather), B = h2 rows.
__global__ void k_gemm3(const float* __restrict__ s, const float* __restrict__ h2,
                        float* __restrict__ out_selu) {
  int lane = threadIdx.x & 31;
  int wave = threadIdx.x >> 5;              // 8 N-tiles (HID=128)
  int half = lane >> 4, idx = lane & 15;
  int g = blockIdx.x >> 1;                  // graph
  int mt = blockIdx.x & 1;                  // M tile (K_=32 -> 2 tiles)
  const float* sg = s + (size_t)g * N_ * K_;
  const float* xg = h2 + (size_t)g * N_ * HID;
  int n0 = wave << 4;
  v8f acc = {};
  for (int kk = 0; kk < N_; kk += 4) {
    int r0 = kk + 2 * half;
    v2f a, b;
    a.x = sg[r0 * K_ + mt * 16 + idx];
    a.y = sg[(r0 + 1) * K_ + mt * 16 + idx];
    b.x = xg[(size_t)r0 * HID + n0 + idx];
    b.y = xg[(size_t)(r0 + 1) * HID + n0 + idx];
    acc = wmma4(a, b, acc);
  }
#pragma unroll
  for (int v = 0; v < 8; ++v) {
    int m = mt * 16 + v + 8 * half;
    out_selu[(size_t)g * K_ * HID + m * HID + n0 + idx] = selu_f(acc[v]);
  }
}

// ---------------- t = A @ s via edge scatter: t[src] += s[dst] ----------------
__global__ void k_tscatter(const int* __restrict__ ei, const float* __restrict__ s,
                           float* __restrict__ t) {
  int tt = blockIdx.x * 256 + threadIdx.x;  // E*K lanes
  int e = tt >> 5;
  int k = tt & 31;
  int sn = ei[e], dn = ei[E_ + e];
  atomicAdd(t + (size_t)sn * K_ + k, s[(size_t)dn * K_ + k]);
}

// ---------------- per-graph stats: trace(s^T t), ca, cluster_size, deg sum ----------------
__global__ void k_gstats(const float* __restrict__ s, const float* __restrict__ t,
                         const float* __restrict__ degcnt, float* __restrict__ trace,
                         float* __restrict__ ca, float* __restrict__ cs,
                         float* __restrict__ degsum) {
  __shared__ float l1[256], l2[256], l3[256], l4[256];
  int b = blockIdx.x;
  int tid = threadIdx.x;
  int k = tid & 31, g = tid >> 5;           // 8 node-groups
  const float* sg = s + (size_t)b * N_ * K_;
  const float* tg = t + (size_t)b * N_ * K_;
  const float* dg = degcnt + (size_t)b * N_;
  float cak = 0.f, csk = 0.f, tr = 0.f, ds = 0.f;
  for (int n = g; n < N_; n += 8) {
    float sv = sg[n * K_ + k];
    float dv = dg[n];
    cak += sv * dv;
    csk += sv;
    tr  += sv * tg[n * K_ + k];
    if (k == 0) ds += dv;
  }
  l1[tid] = cak; l2[tid] = csk; l3[tid] = tr; l4[tid] = ds;
  __syncthreads();
  if (g == 0) {
    float c1 = 0.f, c2 = 0.f;
    for (int j = 0; j < 8; ++j) { c1 += l1[k + 32 * j]; c2 += l2[k + 32 * j]; }
    ca[b * K_ + k] = c1;
    cs[b * K_ + k] = c2;
  }
  for (int st = 128; st > 0; st >>= 1) {
    __syncthreads();
    if (tid < st) { l3[tid] += l3[tid + st]; l4[tid] += l4[tid + st]; }
  }
  __syncthreads();
  if (tid == 0) { trace[b] = l3[0]; degsum[b] = l4[0]; }
}

// ---------------- ss = s^T s per graph (32x32) ----------------
__global__ void k_ss(const float* __restrict__ s, float* __restrict__ ss) {
  __shared__ float rows[16 * K_];           // 16 staged s-rows
  int b = blockIdx.x;
  int tid = threadIdx.x;
  const float* sg = s + (size_t)b * N_ * K_;
  int p0 = tid * 4;                         // 4 (k,l) pairs per thread
  int kk = p0 >> 5, ll0 = p0 & 31;
  float acc[4] = {0.f, 0.f, 0.f, 0.f};
  for (int n0 = 0; n0 < N_; n0 += 16) {
    rows[tid]       = sg[(size_t)n0 * K_ + tid];
    rows[tid + 256] = sg[(size_t)n0 * K_ + 256 + tid];
    __syncthreads();
    for (int r = 0; r < 16; ++r) {
      float a = rows[r * K_ + kk];
#pragma unroll
      for (int j = 0; j < 4; ++j) acc[j] += a * rows[r * K_ + ll0 + j];
    }
    __syncthreads();
  }
#pragma unroll
  for (int j = 0; j < 4; ++j) ss[(size_t)b * K_ * K_ + p0 + j] = acc[j];
}

// ---------------- final losses (tiny: one thread per graph) ----------------
__global__ void k_loss(const float* __restrict__ trace, const float* __restrict__ ca,
                       const float* __restrict__ cs, const float* __restrict__ ss,
                       const float* __restrict__ degsum, float* __restrict__ loss_out) {
  __shared__ float l[16];
  int b = threadIdx.x;
  float v = 0.f;
  if (b < B_) {
    float m = degsum[b] * 0.5f;
    float sum_ca2 = 0.f;
    for (int k = 0; k < K_; ++k) { float c = ca[b * K_ + k]; sum_ca2 += c * c; }
    float spec = -(trace[b] - sum_ca2 / (2.f * m)) / (2.f * m);

    const float* S = ss + (size_t)b * K_ * K_;
    float fro2 = 0.f;
    for (int i = 0; i < K_ * K_; ++i) fro2 += S[i] * S[i];
    float inv = rsqrtf(fro2);
    float isk = rsqrtf((float)K_);
    float d2 = 0.f;
    for (int i = 0; i < K_ * K_; ++i) {
      float dd = S[i] * inv - ((i % (K_ + 1) == 0) ? isk : 0.f);
      d2 += dd * dd;
    }
    float ortho = sqrtf(d2);

    float cs2 = 0.f;
    for (int k = 0; k < K_; ++k) { float c = cs[b * K_ + k]; cs2 += c * c; }
    float clus = sqrtf(cs2) / (float)N_ * sqrtf((float)K_) - 1.f;

    v = (spec + ortho + clus) / (float)B_;
  }
  l[b < 16 ? b : 0] = 0.f;
  __syncthreads();
  if (b < B_) l[b] = v;
  __syncthreads();
  if (b == 0) {
    float tot = 0.f;
    for (int i = 0; i < B_; ++i) tot += l[i];
    loss_out[0] = tot;
  }
}

// ---------------- log_softmax over HID per (b,k) row ----------------
__global__ void k_logsoftmax(const float* __restrict__ out_selu, float* __restrict__ out) {
  __shared__ float red[128];
  int row = blockIdx.x;                     // B*K = 512 rows
  int tid = threadIdx.x;                    // 128
  float v = out_selu[(size_t)row * HID + tid];
  red[tid] = v;
  __syncthreads();
  for (int st = 64; st > 0; st >>= 1) {
    if (tid < st) red[tid] = fmaxf(red[tid], red[tid + st]);
    __syncthreads();
  }
  float mx = red[0];
  __syncthreads();
  red[tid] = expf(v - mx);
  __syncthreads();
  for (int st = 64; st > 0; st >>= 1) {
    if (tid < st) red[tid] += red[tid + st];
    __syncthreads();
  }
  float lsum = logf(red[0]);
  out[(size_t)row * HID + tid] = v - mx - lsum;
}

// ---------------- launcher ----------------
extern "C" void kernel_launch(void* const* d_in, const int* in_sizes, int n_in,
                              void* d_out, int out_size, void* d_ws, size_t ws_size,
                              hipStream_t stream) {
  const float* x   = (const float*)d_in[0];
  const int*   ei  = (const int*)d_in[1];
  // d_in[2] = batch (sorted, uniform sizes -> node n belongs to graph n/N_)
  const float* gw  = (const float*)d_in[3];
  const float* gb  = (const float*)d_in[4];
  const float* gms = (const float*)d_in[5];
  const float* w1  = (const float*)d_in[6];
  const float* b1  = (const float*)d_in[7];
  const float* w2  = (const float*)d_in[8];
  const float* b2  = (const float*)d_in[9];

  float* out = (float*)d_out;
  float* out_lsm  = out;                    // [B,K,HID] = 65536
  float* out_loss = out + 65536;            // scalar
  float* s        = out + 65537;            // [NT,K] = 1048576

  float* ws = (float*)d_ws;
  size_t off = 0;
  float* h       = ws + off; off += (size_t)NT * CIN;   // 2,097,152
  float* hw      = ws + off; off += (size_t)NT * HID;   // 4,194,304
  float* agg     = ws + off; off += (size_t)NT * HID;   // h2 in-place after selu
  float* logits  = ws + off; off += (size_t)NT * K_;
  float* t       = ws + off; off += (size_t)NT * K_;
  float* dinv    = ws + off; off += NT;                 // deg then rsqrt in place
  float* degcnt  = ws + off; off += NT;
  float* mean    = ws + off; off += B_ * CIN;
  float* var     = ws + off; off += B_ * CIN;
  float* outselu = ws + off; off += B_ * K_ * HID;
  float* ssb     = ws + off; off += B_ * K_ * K_;
  float* ca      = ws + off; off += B_ * K_;
  float* cs      = ws + off; off += B_ * K_;
  float* trace   = ws + off; off += B_;
  float* degsum  = ws + off; off += B_;

  // init accumulators (t, deg=1 self-loop, degcnt=0)
  k_init<<<(NT * K_) / 256, 256, 0, stream>>>(t, dinv, degcnt);

  // GraphNorm
  k_mean<<<B_, 256, 0, stream>>>(x, mean);
  k_var<<<B_, 256, 0, stream>>>(x, mean, gms, var);
  k_norm<<<(NT * CIN) / 256, 256, 0, stream>>>(x, mean, var, gms, gw, gb, h);

  // degrees + D^-1/2
  k_degree<<<E_ / 256, 256, 0, stream>>>(ei, dinv, degcnt);
  k_dinv<<<NT / 256, 256, 0, stream>>>(dinv);

  // GCN: hw = h @ w1 (WMMA), then normalized scatter + self loops, bias, selu
  k_gemm1<<<NT / 16, 256, 0, stream>>>(h, w1, hw);
  k_selfloop<<<(NT * HID) / 256, 256, 0, stream>>>(hw, dinv, agg);
  k_edgeagg<<<((size_t)E_ * HID) / 256, 256, 0, stream>>>(ei, hw, dinv, agg);
  k_selu1<<<(NT * HID) / 256, 256, 0, stream>>>(agg, b1);   // agg -> h2

  // cluster assignment: logits = h2 @ w2 + b2 (WMMA), softmax -> s
  k_gemm2<<<NT / 64, 256, 0, stream>>>(agg, w2, b2, logits);
  k_softmax<<<NT / 256, 256, 0, stream>>>(logits, s);

  // out = selu(s^T @ xd) per graph (WMMA)
  k_gemm3<<<B_ * 2, 256, 0, stream>>>(s, agg, outselu);

  // trace(s^T A s) via t = A s edge scatter; per-graph stats
  k_tscatter<<<((size_t)E_ * K_) / 256, 256, 0, stream>>>(ei, s, t);
  k_gstats<<<B_, 256, 0, stream>>>(s, t, degcnt, trace, ca, cs, degsum);
  k_ss<<<B_, 256, 0, stream>>>(s, ssb);
  k_loss<<<1, 16, 0, stream>>>(trace, ca, cs, ssb, degsum, out_loss);

  // log_softmax over feature axis
  k_logsoftmax<<<B_ * K_, 128, 0, stream>>>(outselu, out_lsm);
}